// EmbeddingLayer_44066364457359
// MI455X (gfx1250) — compile-verified
//
#include <hip/hip_runtime.h>

typedef float v2f __attribute__((ext_vector_type(2)));
typedef float v8f __attribute__((ext_vector_type(8)));

#define B_   16
#define C_   128
#define H_   152
#define W_   272
#define G_   8
#define CG_  16
#define S_   4
#define CS_  32
#define BG_  (B_*G_)        // 128
#define HW_  (H_*W_)        // 41344
#define HPW_ (H_+W_)        // 424
#define EPS_ 1e-5f

// LDS staging geometry for the 3x3 conv: 3 rows x 274 cols (halo) x 16 ch,
// column stride 17 floats -> bank-conflict-free for both staging stores and
// the per-lane B-fragment ds loads (17*dcol + dci == 0 mod 64 has no solution
// for |dcol|<=15, dci in {0,+-2}).
#define LCOLS 274
#define LSTR  17

__device__ __forceinline__ float sigf(float v) { return 1.0f / (1.0f + expf(-v)); }

// ---------------- K0: zero the atomic accumulator ----------------
__global__ void k_zero(float* __restrict__ p, int n) {
    int i = blockIdx.x * blockDim.x + threadIdx.x;
    if (i < n) p[i] = 0.0f;
}

// ---------------- K1: per-channel row means + col means into contiguous cat ----------------
__global__ void k_rowcol(const float* __restrict__ x, float* __restrict__ cat) {
    int bgc = blockIdx.x;                 // 0..2047 == global channel b*C + ch
    const float* plane = x + (size_t)bgc * HW_;
    float* cv = cat + (size_t)bgc * HPW_; // [0..151]=row means, [152..423]=col means
    __shared__ float red[256];
    int t = threadIdx.x;
    float cs0 = 0.0f, cs1 = 0.0f;
    for (int h = 0; h < H_; ++h) {
        const float* row = plane + (size_t)h * W_;
        float v0 = row[t];                               // t<256<272 always valid
        float v1 = (t + 256 < W_) ? row[t + 256] : 0.0f;
        cs0 += v0; cs1 += v1;
        red[t] = v0 + v1;
        __syncthreads();
        for (int s = 128; s > 0; s >>= 1) { if (t < s) red[t] += red[t + s]; __syncthreads(); }
        if (t == 0) cv[h] = red[0] * (1.0f / W_);
        __syncthreads();
    }
    cv[H_ + t] = cs0 * (1.0f / H_);
    if (t + 256 < W_) cv[H_ + t + 256] = cs1 * (1.0f / H_);
}

// ---------------- K2: 1x1 conv over cat via WMMA f32 16x16x4, sigmoid -> sig ----------------
__global__ void k_conv1(const float* __restrict__ cat,
                        const float* __restrict__ w1, const float* __restrict__ b1,
                        float* __restrict__ sig) {
    int bg   = blockIdx.x;                 // 128
    int lane = threadIdx.x & 31;
    int wid  = threadIdx.x >> 5;           // 4 waves
    int n    = lane & 15;
    int half = lane >> 4;

    // A fragments: A[co,ci] = w1[co*16+ci]
    float a0[4], a1[4];
#pragma unroll
    for (int cc = 0; cc < 4; ++cc) {
        int k = 4 * cc + 2 * half;
        a0[cc] = w1[n * CG_ + k];
        a1[cc] = w1[n * CG_ + k + 1];
    }
    float bias[8];
#pragma unroll
    for (int r = 0; r < 8; ++r) bias[r] = b1[r + 8 * half];

    for (int tile = wid; tile < 27; tile += 4) {      // ceil(424/16)=27
        int p   = tile * 16 + n;
        int pc  = p < HPW_ ? p : HPW_ - 1;            // clamp, always in-bounds
        float m = p < HPW_ ? 1.0f : 0.0f;
        v8f c;
#pragma unroll
        for (int r = 0; r < 8; ++r) c[r] = bias[r];
#pragma unroll
        for (int cc = 0; cc < 4; ++cc) {
            int k = 4 * cc + 2 * half;
            v2f a; a[0] = a0[cc]; a[1] = a1[cc];
            v2f b;
            b[0] = cat[(size_t)(bg * CG_ + k)     * HPW_ + pc] * m;
            b[1] = cat[(size_t)(bg * CG_ + k + 1) * HPW_ + pc] * m;
            c = __builtin_amdgcn_wmma_f32_16x16x4_f32(false, a, false, b,
                                                      (short)0, c, false, false);
        }
        if (p < HPW_) {
#pragma unroll
            for (int r = 0; r < 8; ++r) {
                int co = r + 8 * half;
                sig[(size_t)(bg * CG_ + co) * HPW_ + p] = sigf(c[r]);
            }
        }
    }
}

// ---------------- K3: per-(bg,cg) sum / sumsq of pre = gx*sig_h*sig_w ----------------
__global__ void k_prestats(const float* __restrict__ x, const float* __restrict__ sig,
                           float* __restrict__ sum_pre, float* __restrict__ sumsq_pre) {
    int bgc = blockIdx.x;                 // 2048
    const float* plane = x + (size_t)bgc * HW_;
    const float* shv = sig + (size_t)bgc * HPW_;       // [h]
    const float* swv = shv + H_;                       // [w]
    int t = threadIdx.x;
    float s = 0.0f, ss = 0.0f;
    for (int i = t; i < HW_; i += 256) {
        int h = i / W_, w = i - h * W_;
        float p = plane[i] * shv[h] * swv[w];
        s += p; ss += p * p;
    }
    __shared__ float r0[256], r1[256];
    r0[t] = s; r1[t] = ss;
    __syncthreads();
    for (int k = 128; k > 0; k >>= 1) {
        if (t < k) { r0[t] += r0[t + k]; r1[t] += r1[t + k]; }
        __syncthreads();
    }
    if (t == 0) { sum_pre[bgc] = r0[0]; sumsq_pre[bgc] = r1[0]; }
}

// ---------------- K4: 3x3 conv (16->16 ch), LDS-staged, WMMA f32 16x16x4 ----------------
__global__ void k_conv3(const float* __restrict__ x,
                        const float* __restrict__ w3, const float* __restrict__ b3,
                        float* __restrict__ x2buf, float* __restrict__ sum_x2) {
    __shared__ float lds[3 * LCOLS * LSTR];            // ~55 KB, zero-padded halo
    int blk  = blockIdx.x;                // 128*152
    int bg   = blk / H_;
    int h    = blk - bg * H_;
    const float* gx = x + (size_t)bg * CG_ * HW_;
    int lane = threadIdx.x & 31;
    int wid  = threadIdx.x >> 5;          // 4 waves
    int n    = lane & 15;
    int half = lane >> 4;

    // Stage rows h-1..h+1, cols -1..272, all 16 channels; zero padding baked in.
    // i -> (r, ci, c): c fastest for coalesced global reads; LDS layout [r][c][ci].
    for (int i = threadIdx.x; i < 3 * CG_ * LCOLS; i += 128) {
        int r   = i / (CG_ * LCOLS);
        int rem = i - r * (CG_ * LCOLS);
        int ci  = rem / LCOLS;
        int c   = rem - ci * LCOLS;
        int hh  = h + r - 1;
        int gc  = c - 1;
        float msk = ((unsigned)hh < (unsigned)H_ && (unsigned)gc < (unsigned)W_) ? 1.0f : 0.0f;
        int hc = hh < 0 ? 0 : (hh >= H_ ? H_ - 1 : hh);
        int cc = gc < 0 ? 0 : (gc >= W_ ? W_ - 1 : gc);
        lds[(r * LCOLS + c) * LSTR + ci] = gx[(size_t)ci * HW_ + (size_t)hc * W_ + cc] * msk;
    }

    // Preload A fragments for 9 taps x 4 K-chunks: A[co,ci] = w3[(co*16+ci)*9 + tap]
    float a0[36], a1[36];
#pragma unroll
    for (int tap = 0; tap < 9; ++tap)
#pragma unroll
        for (int cc = 0; cc < 4; ++cc) {
            int k = 4 * cc + 2 * half;
            a0[tap * 4 + cc] = w3[(size_t)(n * CG_ + k)     * 9 + tap];
            a1[tap * 4 + cc] = w3[(size_t)(n * CG_ + k + 1) * 9 + tap];
        }
    float bias[8];
#pragma unroll
    for (int r = 0; r < 8; ++r) bias[r] = b3[r + 8 * half];

    float accsum[8];
#pragma unroll
    for (int r = 0; r < 8; ++r) accsum[r] = 0.0f;

    __syncthreads();

    for (int tile = wid; tile < 17; tile += 4) {      // 272/16 = 17
        int x0 = tile * 16;
        v8f c;
#pragma unroll
        for (int r = 0; r < 8; ++r) c[r] = bias[r];
#pragma unroll
        for (int tap = 0; tap < 9; ++tap) {
            int dy = tap / 3, dx = tap % 3;           // 0..2, LDS already shifted by halo
            int base = (dy * LCOLS + (x0 + n + dx)) * LSTR;
            // batch the 8 LDS reads for this tap, then 4 chained WMMAs
            float bb0[4], bb1[4];
#pragma unroll
            for (int cc = 0; cc < 4; ++cc) {
                int k = 4 * cc + 2 * half;
                bb0[cc] = lds[base + k];
                bb1[cc] = lds[base + k + 1];
            }
#pragma unroll
            for (int cc = 0; cc < 4; ++cc) {
                v2f a; a[0] = a0[tap * 4 + cc]; a[1] = a1[tap * 4 + cc];
                v2f b; b[0] = bb0[cc];          b[1] = bb1[cc];
                c = __builtin_amdgcn_wmma_f32_16x16x4_f32(false, a, false, b,
                                                          (short)0, c, false, false);
            }
        }
        float* outb = x2buf + (size_t)bg * CG_ * HW_ + (size_t)h * W_ + x0 + n;
#pragma unroll
        for (int r = 0; r < 8; ++r) {
            int co = r + 8 * half;
            outb[(size_t)co * HW_] = c[r];
            accsum[r] += c[r];
        }
    }
    // reduce over the 16 "N" lanes inside each half, then one atomicAdd per co
#pragma unroll
    for (int r = 0; r < 8; ++r) {
        float v = accsum[r];
        for (int m = 1; m < 16; m <<= 1) v += __shfl_xor(v, m, 32);  // stays within half
        if (n == 0) atomicAdd(&sum_x2[bg * CG_ + r + 8 * half], v);
    }
}

// ---------------- K5: per-bg alpha/beta + global x11 = softmax(gn_b) ----------------
__global__ void k_prep(const float* __restrict__ gn_w, const float* __restrict__ gn_b,
                       const float* __restrict__ sum_pre, const float* __restrict__ sumsq_pre,
                       const float* __restrict__ sum_x2,
                       float* __restrict__ alpha, float* __restrict__ beta,
                       float* __restrict__ x11v) {
    int bg = blockIdx.x;                  // 128
    int t  = threadIdx.x;                 // 32
    __shared__ float mu[16], rs[16], x2m[16];
    if (t < 16) {
        float m = sum_pre[bg * 16 + t] * (1.0f / HW_);
        float var = sumsq_pre[bg * 16 + t] * (1.0f / HW_) - m * m;
        mu[t] = m;
        rs[t] = rsqrtf(var + EPS_);
        x2m[t] = sum_x2[bg * 16 + t] * (1.0f / HW_);
    }
    __syncthreads();
    if (t == 0) {
        float mx = x2m[0];
        for (int i = 1; i < 16; ++i) mx = fmaxf(mx, x2m[i]);
        float e[16], den = 0.0f;
        for (int i = 0; i < 16; ++i) { e[i] = expf(x2m[i] - mx); den += e[i]; }
        float inv = 1.0f / den, bsum = 0.0f;
        for (int i = 0; i < 16; ++i) {
            float x21 = e[i] * inv;
            alpha[bg * 16 + i] = x21 * gn_w[i] * rs[i];
            bsum += x21 * (gn_b[i] - gn_w[i] * mu[i] * rs[i]);
        }
        beta[bg] = bsum;
        if (bg == 0) {            // x1.mean((2,3)) == gn_b exactly -> x11 = softmax(gn_b)
            float m2 = gn_b[0];
            for (int i = 1; i < 16; ++i) m2 = fmaxf(m2, gn_b[i]);
            float e2[16], d2 = 0.0f;
            for (int i = 0; i < 16; ++i) { e2[i] = expf(gn_b[i] - m2); d2 += e2[i]; }
            for (int i = 0; i < 16; ++i) x11v[i] = e2[i] / d2;
        }
    }
}

// ---------------- K6: weights + spc = gx*sigmoid(weights), in-place over x2 ----------------
__global__ void k_spc(const float* __restrict__ x, const float* __restrict__ sig,
                      const float* __restrict__ alpha, const float* __restrict__ beta,
                      const float* __restrict__ x11v,
                      float* __restrict__ spc /* holds x2, overwritten */) {
    int blk = blockIdx.x;                 // 128*152
    int bg  = blk / H_;
    int h   = blk - bg * H_;
    const float* gx = x   + (size_t)bg * CG_ * HW_ + (size_t)h * W_;
    float*       xp = spc + (size_t)bg * CG_ * HW_ + (size_t)h * W_;
    __shared__ float a_s[16], x11_s[16], sh_s[16];
    __shared__ float beta_s;
    int t = threadIdx.x;
    if (t < 16) {
        a_s[t]   = alpha[bg * 16 + t];
        x11_s[t] = x11v[t];
        sh_s[t]  = sig[(size_t)(bg * 16 + t) * HPW_ + h];
    }
    if (t == 0) beta_s = beta[bg];
    __syncthreads();
    for (int w = t; w < W_; w += blockDim.x) {
        float gxr[16];
        float wt = beta_s;
#pragma unroll
        for (int co = 0; co < 16; ++co) {
            float g = gx[(size_t)co * HW_ + w];
            gxr[co] = g;
            float pre = g * sh_s[co] * sig[(size_t)(bg * 16 + co) * HPW_ + H_ + w];
            float x2v = xp[(size_t)co * HW_ + w];
            wt += x11_s[co] * x2v + a_s[co] * pre;
        }
        float sg = sigf(wt);
#pragma unroll
        for (int co = 0; co < 16; ++co)
            xp[(size_t)co * HW_ + w] = gxr[co] * sg;
    }
}

// ---------------- K7: detection-region small convs, in place on spc ----------------
__global__ void k_det(const int* __restrict__ det,
                      const float* __restrict__ cw0, const float* __restrict__ cb0,
                      const float* __restrict__ cw1, const float* __restrict__ cb1,
                      const float* __restrict__ cw2, const float* __restrict__ cb2,
                      const float* __restrict__ cw3, const float* __restrict__ cb3,
                      float* __restrict__ spc) {
    int b   = blockIdx.x >> 2;
    int idx = blockIdx.x & 3;
    int cx = det[(b * S_ + idx) * 4 + 0];
    int cy = det[(b * S_ + idx) * 4 + 1];
    if (!(cy >= 2 && cy < H_ - 2 && cx >= 2 && cx < W_ - 2)) return;
    int ys = cy - 2, xs = cx - 2;
    const float* cw; const float* cb;
    if      (idx == 0) { cw = cw0; cb = cb0; }
    else if (idx == 1) { cw = cw1; cb = cb1; }
    else if (idx == 2) { cw = cw2; cb = cb2; }
    else               { cw = cw3; cb = cb3; }
    int p = idx + 1, k = 2 * p + 1;

    __shared__ float reg[CS_][5][5];
    __shared__ float outp[CS_][5][5];
    float* slab = spc + (size_t)(b * C_ + idx * CS_) * HW_;
    int t = threadIdx.x;
    for (int i = t; i < CS_ * 25; i += blockDim.x) {
        int ci = i / 25, rem = i - ci * 25, y = rem / 5, xx = rem - y * 5;
        reg[ci][y][xx] = slab[(size_t)ci * HW_ + (size_t)(ys + y) * W_ + (xs + xx)];
    }
    __syncthreads();
    for (int i = t; i < CS_ * 25; i += blockDim.x) {
        int co = i / 25, rem = i - co * 25, y = rem / 5, xx = rem - y * 5;
        float acc = cb[co];
        for (int ci = 0; ci < CS_; ++ci)
            for (int ky = 0; ky < k; ++ky) {
                int yy = y + ky - p;
                if ((unsigned)yy >= 5u) continue;
                for (int kx = 0; kx < k; ++kx) {
                    int xw2 = xx + kx - p;
                    if ((unsigned)xw2 >= 5u) continue;
                    acc += cw[((size_t)(co * CS_ + ci) * k + ky) * k + kx] * reg[ci][yy][xw2];
                }
            }
        outp[co][y][xx] = acc;
    }
    __syncthreads();
    for (int i = t; i < CS_ * 25; i += blockDim.x) {
        int ci = i / 25, rem = i - ci * 25, y = rem / 5, xx = rem - y * 5;
        slab[(size_t)ci * HW_ + (size_t)(ys + y) * W_ + (xs + xx)] = outp[ci][y][xx];
    }
}

// ---------------- K8: out = (x + silu(spc)) with channel shuffle ----------------
__global__ void k_final(const float* __restrict__ x, const float* __restrict__ spc,
                        float* __restrict__ out) {
    size_t total  = (size_t)B_ * C_ * HW_;
    size_t stride = (size_t)gridDim.x * blockDim.x;
    for (size_t i = (size_t)blockIdx.x * blockDim.x + threadIdx.x; i < total; i += stride) {
        size_t pos = i % HW_;
        size_t ci  = (i / HW_) % C_;
        size_t b   = i / ((size_t)C_ * HW_);
        int g  = (int)(ci >> 4);
        int cg = (int)(ci & 15);
        int co = cg * G_ + g;
        float v = spc[i];
        out[(b * C_ + co) * HW_ + pos] = x[i] + v * sigf(v);
    }
}

extern "C" void kernel_launch(void* const* d_in, const int* in_sizes, int n_in,
                              void* d_out, int out_size, void* d_ws, size_t ws_size,
                              hipStream_t stream) {
    const float* x    = (const float*)d_in[0];
    const int*   det  = (const int*)  d_in[1];
    const float* w1   = (const float*)d_in[2];
    const float* b1   = (const float*)d_in[3];
    const float* w3   = (const float*)d_in[4];
    const float* b3   = (const float*)d_in[5];
    const float* gn_w = (const float*)d_in[6];
    const float* gn_b = (const float*)d_in[7];
    const float* cw0  = (const float*)d_in[8];
    const float* cb0  = (const float*)d_in[9];
    const float* cw1  = (const float*)d_in[10];
    const float* cb1  = (const float*)d_in[11];
    const float* cw2  = (const float*)d_in[12];
    const float* cb2  = (const float*)d_in[13];
    const float* cw3  = (const float*)d_in[14];
    const float* cb3  = (const float*)d_in[15];
    float* out = (float*)d_out;

    float* ws = (float*)d_ws;
    size_t o = 0;
    float* cat       = ws + o; o += (size_t)BG_ * CG_ * HPW_;  // [xh | xw] means
    float* sigbuf    = ws + o; o += (size_t)BG_ * CG_ * HPW_;  // sigmoid gates
    float* sum_pre   = ws + o; o += BG_ * CG_;
    float* sumsq_pre = ws + o; o += BG_ * CG_;
    float* sum_x2    = ws + o; o += BG_ * CG_;
    float* alpha     = ws + o; o += BG_ * CG_;
    float* betav     = ws + o; o += 128;
    float* x11v      = ws + o; o += 64;
    float* x2buf     = ws + o; o += (size_t)BG_ * CG_ * HW_;   // x2, then spc (in place)

    k_zero    <<<dim3(8),         dim3(256), 0, stream>>>(sum_x2, BG_ * CG_);
    k_rowcol  <<<dim3(BG_ * CG_), dim3(256), 0, stream>>>(x, cat);
    k_conv1   <<<dim3(BG_),       dim3(128), 0, stream>>>(cat, w1, b1, sigbuf);
    k_prestats<<<dim3(BG_ * CG_), dim3(256), 0, stream>>>(x, sigbuf, sum_pre, sumsq_pre);
    k_conv3   <<<dim3(BG_ * H_),  dim3(128), 0, stream>>>(x, w3, b3, x2buf, sum_x2);
    k_prep    <<<dim3(BG_),       dim3(32),  0, stream>>>(gn_w, gn_b, sum_pre, sumsq_pre,
                                                          sum_x2, alpha, betav, x11v);
    k_spc     <<<dim3(BG_ * H_),  dim3(256), 0, stream>>>(x, sigbuf, alpha, betav, x11v, x2buf);
    k_det     <<<dim3(B_ * S_),   dim3(256), 0, stream>>>(det, cw0, cb0, cw1, cb1,
                                                          cw2, cb2, cw3, cb3, x2buf);
    k_final   <<<dim3(16384),     dim3(256), 0, stream>>>(x, x2buf, out);
}